// GCN_79877801771412
// MI455X (gfx1250) — compile-verified
//
#include <hip/hip_runtime.h>

typedef __attribute__((ext_vector_type(2))) float v2f;
typedef __attribute__((ext_vector_type(8))) float v8f;

#define G_DIM   128
#define H1      128
#define H2      128
#define NREL    8
#define NBASES  30
#define NN      50000
#define NE      800000
#define WCOLS   1152            // 8*128 relation outputs + 128 self-loop outputs
#define MTILES  (NN / 16)       // 3125 (exact)
#define NGRPS   (WCOLS / 128)   // 9
#define MBLKS   ((MTILES + 7) / 8)  // 391 blocks of 8 M-tiles
#define LDS_STRIDE 132          // padded row stride: bank-conflict-free b64 fragment reads

// ---------------- K0: zero scratch (agg / agg2 / degrees) ----------------
__global__ void k_zero_f4(float4* __restrict__ p, int n4) {
    int i = blockIdx.x * blockDim.x + threadIdx.x;
    if (i < n4) p[i] = make_float4(0.f, 0.f, 0.f, 0.f);
}

// ---------------- K1: Wt[c][i] = combined weight, K-contiguous ----------------
// c in [0,1024): relation r=c>>7, out o=c&127 : sum_b w_comp[r,b]*bases[b,i,o]
// c in [1024,1152): self-loop column (c-1024) : loop_weight[i, c-1024]
__global__ void k_build_wt(const float* __restrict__ w_comp,
                           const float* __restrict__ bases,
                           const float* __restrict__ loop_w,
                           float* __restrict__ Wt) {
    int idx = blockIdx.x * blockDim.x + threadIdx.x;   // c*128 + i
    if (idx >= WCOLS * G_DIM) return;
    int c = idx >> 7;
    int i = idx & 127;
    float v;
    if (c < NREL * H1) {
        int r = c >> 7;
        int o = c & 127;
        float acc = 0.f;
#pragma unroll
        for (int b = 0; b < NBASES; ++b)
            acc = fmaf(w_comp[r * NBASES + b], bases[((size_t)b * G_DIM + i) * H1 + o], acc);
        v = acc;
    } else {
        v = loop_w[(size_t)i * H1 + (c - NREL * H1)];
    }
    Wt[(size_t)c * G_DIM + i] = v;
}

// ---------------- K2: out1[N,1152] = X[N,128] @ Wcat[128,1152]  (WMMA f32) ----
// Block = 8 waves sharing one 128-col N-group; B panel staged in LDS once.
// Each wave computes a 16x128 slab: 8 f32 WMMA accumulators, 8x A reuse.
__global__ void __launch_bounds__(256)
k_gemm_xw(const float* __restrict__ X,
          const float* __restrict__ Wt,     // [1152,128] K-contiguous
          float* __restrict__ out1) {
    __shared__ float Bs[128 * LDS_STRIDE];  // 67.6 KB of 320 KB WGP LDS

    int mblk = blockIdx.x / NGRPS;
    int ngrp = blockIdx.x % NGRPS;
    int n0 = ngrp * 128;

    // ---- cooperative stage of Wt[n0..n0+127][0..127] into LDS (float4) ----
    for (int it = 0; it < 16; ++it) {
        int idx4 = threadIdx.x + it * 256;          // 4096 float4 total
        int col  = idx4 >> 5;                       // 32 float4 per row
        int kk4  = idx4 & 31;
        float4 v = *((const float4*)(Wt + (size_t)(n0 + col) * G_DIM) + kk4);
        *((float4*)(&Bs[col * LDS_STRIDE + kk4 * 4])) = v;   // stride 132 ≡ 0 mod 4
    }
    __syncthreads();

    int mtile = mblk * 8 + (threadIdx.x >> 5);
    if (mtile >= MTILES) return;                    // wave-uniform; EXEC all-1 below
    int m0 = mtile * 16;
    int lane = threadIdx.x & 31;
    int half = lane >> 4;                           // 0/1
    int mrow = lane & 15;

    const float* arow = X + (size_t)(m0 + mrow) * G_DIM;
    v8f acc[8] = {};

    for (int k0 = 0; k0 < G_DIM; k0 += 4) {
        // A 16x4 fragment: VGPR0/1 = K={0,1} (lanes 0-15), K={2,3} (lanes 16-31)
        v2f a = *(const v2f*)(arow + k0 + 2 * half);
#pragma unroll
        for (int t = 0; t < 8; ++t) {
            // B 4x16 fragment from LDS: banks 4*mrow + k0 + 2*half (+1) -> conflict-free
            v2f b = *(const v2f*)(&Bs[(t * 16 + mrow) * LDS_STRIDE + k0 + 2 * half]);
            acc[t] = __builtin_amdgcn_wmma_f32_16x16x4_f32(
                false, a, false, b, (short)0, acc[t], false, false);
        }
    }
#pragma unroll
    for (int t = 0; t < 8; ++t) {
        int col = n0 + t * 16 + mrow;
#pragma unroll
        for (int v = 0; v < 8; ++v) {
            int row = m0 + v + 8 * half;            // C VGPR v = rows v, v+8
            out1[(size_t)row * WCOLS + col] = acc[t][v];
        }
    }
}

// ---------------- K3: RGCN edge pass: gather xw[et,src], scatter-add to agg[dst]
// 32 lanes per edge, float4 per lane -> one coalesced 512B gather per edge.
__global__ void k_edge_rgcn(const int* __restrict__ ei,     // [2,E]
                            const float* __restrict__ enorm,
                            const int* __restrict__ etype,
                            const float* __restrict__ out1,
                            float* __restrict__ agg,
                            float* __restrict__ outdeg,
                            float* __restrict__ indeg) {
    int t = blockIdx.x * blockDim.x + threadIdx.x;
    int e = t >> 5;
    int lane = t & 31;
    if (e >= NE) return;
    int src = ei[e];
    int dst = ei[NE + e];
    float w = enorm[e];
    int et = etype[e];
    float4 v = *((const float4*)(out1 + (size_t)src * WCOLS + (size_t)et * H1) + lane);
    float* d = agg + (size_t)dst * H1 + lane * 4;
    unsafeAtomicAdd(d + 0, v.x * w);
    unsafeAtomicAdd(d + 1, v.y * w);
    unsafeAtomicAdd(d + 2, v.z * w);
    unsafeAtomicAdd(d + 3, v.w * w);
    if (lane == 0) {
        unsafeAtomicAdd(outdeg + src, 1.f);
        unsafeAtomicAdd(indeg + dst, 1.f);
    }
}

// ---------------- K4: hs = (agg + self-loop + bias1) * outdeg^-1/2 ----------------
__global__ void k_make_hs(const float* __restrict__ agg,
                          const float* __restrict__ out1,
                          const float* __restrict__ bias1,
                          const float* __restrict__ outdeg,
                          float* __restrict__ hs) {
    int idx = blockIdx.x * blockDim.x + threadIdx.x;
    if (idx >= NN * H1) return;
    int n = idx >> 7;
    int o = idx & 127;
    float h = agg[idx] + out1[(size_t)n * WCOLS + NREL * H1 + o] + bias1[o];
    float ns = rsqrtf(fmaxf(outdeg[n], 1.f));
    hs[idx] = h * ns;
}

// ---------------- K5: GraphConv edge pass ----------------
__global__ void k_edge_gcn(const int* __restrict__ ei,
                           const float* __restrict__ enorm,
                           const float* __restrict__ hs,
                           float* __restrict__ agg2) {
    int t = blockIdx.x * blockDim.x + threadIdx.x;
    int e = t >> 5;
    int lane = t & 31;
    if (e >= NE) return;
    int src = ei[e];
    int dst = ei[NE + e];
    float w = enorm[e];
    float4 v = *((const float4*)(hs + (size_t)src * H1) + lane);
    float* d = agg2 + (size_t)dst * H1 + lane * 4;
    unsafeAtomicAdd(d + 0, v.x * w);
    unsafeAtomicAdd(d + 1, v.y * w);
    unsafeAtomicAdd(d + 2, v.z * w);
    unsafeAtomicAdd(d + 3, v.w * w);
}

// ---------------- K6: scale by indeg^-1/2 ----------------
__global__ void k_scale_dst(const float* __restrict__ agg2,
                            const float* __restrict__ indeg,
                            float* __restrict__ agg2s) {
    int idx = blockIdx.x * blockDim.x + threadIdx.x;
    if (idx >= NN * H1) return;
    agg2s[idx] = agg2[idx] * rsqrtf(fmaxf(indeg[idx >> 7], 1.f));
}

// ---------------- K7: out[N,128] = agg2s @ W2 + bias2  (WMMA f32) ----------------
// W2 staged into LDS *transposed* so fragment reads are contiguous b64s.
__global__ void __launch_bounds__(256)
k_gemm_out(const float* __restrict__ A,     // [N,128]
           const float* __restrict__ W2,    // [128,128] row-major
           const float* __restrict__ bias2,
           float* __restrict__ out) {
    __shared__ float Ws[128 * LDS_STRIDE];

    // stage transposed: Ws[c][k] = W2[k][c]; coalesced global reads
    for (int it = 0; it < 64; ++it) {
        int idx = threadIdx.x + it * 256;           // 16384 scalars
        int k = idx >> 7;
        int c = idx & 127;
        Ws[c * LDS_STRIDE + k] = W2[(size_t)k * H2 + c];
    }
    __syncthreads();

    int mtile = blockIdx.x * 8 + (threadIdx.x >> 5);
    if (mtile >= MTILES) return;
    int m0 = mtile * 16;
    int lane = threadIdx.x & 31;
    int half = lane >> 4;
    int mrow = lane & 15;

    const float* arow = A + (size_t)(m0 + mrow) * H1;
    v8f acc[8] = {};

    for (int k0 = 0; k0 < H1; k0 += 4) {
        v2f a = *(const v2f*)(arow + k0 + 2 * half);
#pragma unroll
        for (int t = 0; t < 8; ++t) {
            v2f b = *(const v2f*)(&Ws[(t * 16 + mrow) * LDS_STRIDE + k0 + 2 * half]);
            acc[t] = __builtin_amdgcn_wmma_f32_16x16x4_f32(
                false, a, false, b, (short)0, acc[t], false, false);
        }
    }
#pragma unroll
    for (int t = 0; t < 8; ++t) {
        int col = t * 16 + mrow;
        float bb = bias2[col];
#pragma unroll
        for (int v = 0; v < 8; ++v) {
            int row = m0 + v + 8 * half;
            out[(size_t)row * H2 + col] = acc[t][v] + bb;
        }
    }
}

extern "C" void kernel_launch(void* const* d_in, const int* in_sizes, int n_in,
                              void* d_out, int out_size, void* d_ws, size_t ws_size,
                              hipStream_t stream) {
    const float* X      = (const float*)d_in[0];
    const int*   ei     = (const int*)  d_in[1];
    const float* enorm  = (const float*)d_in[2];
    const int*   etype  = (const int*)  d_in[3];
    const float* w_comp = (const float*)d_in[4];
    const float* bases  = (const float*)d_in[5];
    const float* loop_w = (const float*)d_in[6];
    const float* bias1  = (const float*)d_in[7];
    const float* W2     = (const float*)d_in[8];
    const float* bias2  = (const float*)d_in[9];
    float* out = (float*)d_out;

    // workspace layout (floats)
    float* ws = (float*)d_ws;
    size_t off = 0;
    float* Wt     = ws + off; off += (size_t)WCOLS * G_DIM;   // 147456
    float* out1   = ws + off; off += (size_t)NN * WCOLS;      // 57.6M
    float* agg    = ws + off; off += (size_t)NN * H1;         // zeroed region starts here
    float* agg2   = ws + off; off += (size_t)NN * H1;
    float* outdeg = ws + off; off += NN;
    float* indeg  = ws + off; off += NN;
    float* hs     = ws + off; off += (size_t)NN * H1;
    float* agg2s  = agg;      // agg is dead after K4; reuse for the scaled buffer

    // K0: zero agg | agg2 | outdeg | indeg (contiguous): 12,900,000 floats
    {
        int n4 = (2 * NN * H1 + 2 * NN) / 4;                  // 3,225,000
        k_zero_f4<<<(n4 + 255) / 256, 256, 0, stream>>>((float4*)agg, n4);
    }
    // K1: combined relation+loop weight, transposed (K-contiguous)
    k_build_wt<<<(WCOLS * G_DIM) / 256, 256, 0, stream>>>(w_comp, bases, loop_w, Wt);
    // K2: big projection GEMM (WMMA, LDS-staged B panel)
    k_gemm_xw<<<MBLKS * NGRPS, 256, 0, stream>>>(X, Wt, out1);
    // K3: RGCN message pass + degree counts
    k_edge_rgcn<<<(NE * 32) / 256, 256, 0, stream>>>(ei, enorm, etype, out1,
                                                     agg, outdeg, indeg);
    // K4: h -> hs
    k_make_hs<<<(NN * H1) / 256, 256, 0, stream>>>(agg, out1, bias1, outdeg, hs);
    // K5: GraphConv message pass
    k_edge_gcn<<<(NE * 32) / 256, 256, 0, stream>>>(ei, enorm, hs, agg2);
    // K6: dst-norm scaling
    k_scale_dst<<<(NN * H1) / 256, 256, 0, stream>>>(agg2, indeg, agg2s);
    // K7: output GEMM (WMMA, LDS-staged transposed W2) + bias
    k_gemm_out<<<MBLKS, 256, 0, stream>>>(agg2s, W2, bias2, out);

    (void)in_sizes; (void)n_in; (void)out_size; (void)ws_size;
}